// NonMaxSuppression_15642270892889
// MI455X (gfx1250) — compile-verified
//
#include <hip/hip_runtime.h>

// Problem constants (from reference setup_inputs)
#define B_    16
#define H_    1536
#define W_    2048
#define TOPK  4096
#define TW    128              // tile width  (output pixels)
#define TH    32               // tile height (output pixels)
#define LW    136              // TW + 4-left + 4-right halo (16B-aligned chunks)
#define LH    36               // TH + 2 + 2 halo
#define ROWW  (W_ / 32)        // bitmap words per image row (64)
#define REP_THR 0.1f

// ---------------------------------------------------------------------------
// Pass 1: 5x5 max-pool + mask. Async global->LDS staging (CDNA5 ASYNCcnt path),
// separable max, wave32 ballot -> bitmap word + per-row keypoint counts.
// ---------------------------------------------------------------------------
__global__ __launch_bounds__(256) void nms_pass1(const float* __restrict__ score,
                                                 unsigned* __restrict__ bitmap,
                                                 unsigned* __restrict__ rowCount) {
    __shared__ __align__(16) float tile[LH * LW];   // raw scores (with halo)
    __shared__ float hbuf[LH * TW];                 // horizontal 5-max

    const int t   = threadIdx.x;
    const int tx  = blockIdx.x, ty = blockIdx.y, b = blockIdx.z;
    const int gx0 = tx * TW, gy0 = ty * TH;
    const float* img = score + (size_t)b * H_ * W_;

    // Interior blocks: whole halo in-bounds -> skip -inf prefill.
    const bool interior = (gx0 >= 4) && (gx0 + TW + 4 <= W_) &&
                          (gy0 >= 2) && (gy0 + TH + 2 <= H_);
    if (!interior) {
        const float ninf = -__builtin_inff();
        for (int e = t; e < LH * LW; e += 256) tile[e] = ninf;
        __syncthreads();  // LDS prefill visible before async writes land
    }

    // LDS byte offset of tile[] (low 32 bits of generic pointer == LDS offset)
    const unsigned ldsbase = (unsigned)(unsigned long long)&tile[0];

    // 34 aligned 16-byte chunks per LDS row; chunks are all-in or all-out of
    // bounds horizontally because gx0-4 and W are multiples of 4.
    for (int e = t; e < LH * 34; e += 256) {
        const int r  = e / 34;
        const int ci = e - r * 34;
        const int gy = gy0 - 2 + r;
        const int gx = gx0 - 4 + ci * 4;
        if ((unsigned)gy < (unsigned)H_ && (unsigned)gx < (unsigned)W_) {
            const float* gp = img + (size_t)gy * W_ + gx;
            const unsigned loff = ldsbase + (unsigned)((r * LW + ci * 4) * 4);
            asm volatile("global_load_async_to_lds_b128 %0, %1, off"
                         :: "v"(loff), "v"(gp) : "memory");
        }
    }
    asm volatile("s_wait_asynccnt 0x0" ::: "memory");
    __syncthreads();

    // Horizontal 5-wide max: output col j needs tile cols j+2 .. j+6
    for (int e = t; e < LH * TW; e += 256) {
        const int r = e >> 7;          // / TW
        const int c = e & (TW - 1);
        const float* rp = &tile[r * LW + c + 2];
        hbuf[e] = fmaxf(fmaxf(fmaxf(rp[0], rp[1]), fmaxf(rp[2], rp[3])), rp[4]);
    }
    __syncthreads();

    // Vertical 5-max + mask; each wave covers 32 consecutive columns of one
    // row, so __ballot() is exactly one bitmap word in ascending-column order.
    const unsigned lane = t & 31;
    for (int k = 0; k < 16; ++k) {
        const int p = k * 256 + t;       // 0..4095 tile-local pixel
        const int r = p >> 7;
        const int c = p & (TW - 1);
        const float s = tile[(r + 2) * LW + (c + 4)];
        float vm = hbuf[r * TW + c];
        vm = fmaxf(vm, hbuf[(r + 1) * TW + c]);
        vm = fmaxf(vm, hbuf[(r + 2) * TW + c]);
        vm = fmaxf(vm, hbuf[(r + 3) * TW + c]);
        vm = fmaxf(vm, hbuf[(r + 4) * TW + c]);
        const bool kp = (s == vm) && (s > REP_THR);
        const unsigned word = (unsigned)__ballot(kp);
        if (lane == 0) {
            const int gy = gy0 + r;
            bitmap[(size_t)(b * H_ + gy) * ROWW + ((gx0 + c) >> 5)] = word;
            if (word) atomicAdd(&rowCount[b * H_ + gy], (unsigned)__popc(word));
        }
    }
}

// ---------------------------------------------------------------------------
// Pass 2: per-batch exclusive prefix sum over 1536 row counts (6 per thread).
// ---------------------------------------------------------------------------
__global__ __launch_bounds__(256) void nms_scan(const unsigned* __restrict__ rowCount,
                                                unsigned* __restrict__ rowPrefix) {
    const int b = blockIdx.x;
    const unsigned* rc = rowCount + b * H_;
    unsigned* rp = rowPrefix + b * H_;
    const int t = threadIdx.x;

    unsigned v[6];
    unsigned s = 0;
#pragma unroll
    for (int i = 0; i < 6; ++i) { v[i] = rc[t * 6 + i]; s += v[i]; }

    const unsigned lane = t & 31, wave = (unsigned)t >> 5;
    unsigned incl = s;
#pragma unroll
    for (int d = 1; d < 32; d <<= 1) {
        unsigned n = __shfl_up(incl, d, 32);
        if (lane >= (unsigned)d) incl += n;
    }
    __shared__ unsigned wsum[8];
    if (lane == 31) wsum[wave] = incl;
    __syncthreads();
    unsigned woff = 0;
    for (unsigned w = 0; w < wave; ++w) woff += wsum[w];

    unsigned excl = woff + incl - s;
#pragma unroll
    for (int i = 0; i < 6; ++i) { rp[t * 6 + i] = excl; excl += v[i]; }
}

// ---------------------------------------------------------------------------
// Pass 3: one wave per image row; rank set bits and scatter (x,y) pairs.
// Rows whose base offset already exceeds TOPK exit immediately (most do).
// ---------------------------------------------------------------------------
__global__ __launch_bounds__(32) void nms_emit(const unsigned* __restrict__ bitmap,
                                               const unsigned* __restrict__ rowPrefix,
                                               int* __restrict__ out) {
    const int row = blockIdx.x;
    const int b   = blockIdx.y;
    const unsigned base = rowPrefix[b * H_ + row];
    if (base >= TOPK) return;

    const unsigned lane = threadIdx.x;
    const unsigned* wr = bitmap + (size_t)(b * H_ + row) * ROWW;
    const unsigned w0 = wr[2 * lane];
    const unsigned w1 = wr[2 * lane + 1];
    const unsigned cnt = (unsigned)(__popc(w0) + __popc(w1));

    unsigned incl = cnt;
#pragma unroll
    for (int d = 1; d < 32; d <<= 1) {
        unsigned n = __shfl_up(incl, d, 32);
        if (lane >= (unsigned)d) incl += n;
    }
    unsigned k = base + incl - cnt;      // rank of first keypoint in my words

    int colbase = (int)lane * 64;
    unsigned m = w0;
    while (m) {
        const int bit = __ffs(m) - 1; m &= m - 1;
        if (k < TOPK) {
            int* o = out + ((size_t)b * TOPK + k) * 2;
            o[0] = colbase + bit;        // x = w
            o[1] = row;                  // y = h
        }
        ++k;
    }
    m = w1; colbase += 32;
    while (m) {
        const int bit = __ffs(m) - 1; m &= m - 1;
        if (k < TOPK) {
            int* o = out + ((size_t)b * TOPK + k) * 2;
            o[0] = colbase + bit;
            o[1] = row;
        }
        ++k;
    }
}

// ---------------------------------------------------------------------------
extern "C" void kernel_launch(void* const* d_in, const int* in_sizes, int n_in,
                              void* d_out, int out_size, void* d_ws, size_t ws_size,
                              hipStream_t stream) {
    (void)in_sizes; (void)n_in; (void)ws_size;
    const float* score = (const float*)d_in[0];
    int* out = (int*)d_out;

    unsigned char* ws = (unsigned char*)d_ws;
    const size_t bitmapBytes = (size_t)B_ * H_ * ROWW * sizeof(unsigned); // 6.29 MB
    const size_t rowsBytes   = (size_t)B_ * H_ * sizeof(unsigned);       // 96 KB
    unsigned* bitmap    = (unsigned*)(ws);
    unsigned* rowCount  = (unsigned*)(ws + bitmapBytes);
    unsigned* rowPrefix = (unsigned*)(ws + bitmapBytes + rowsBytes);

    hipMemsetAsync(rowCount, 0, rowsBytes, stream);
    hipMemsetAsync(d_out, 0, (size_t)out_size * sizeof(int), stream);

    nms_pass1<<<dim3(W_ / TW, H_ / TH, B_), 256, 0, stream>>>(score, bitmap, rowCount);
    nms_scan <<<dim3(B_), 256, 0, stream>>>(rowCount, rowPrefix);
    nms_emit <<<dim3(H_, B_), 32, 0, stream>>>(bitmap, rowPrefix, out);
}